// LightGCN_11261404250192
// MI455X (gfx1250) — compile-verified
//
#include <hip/hip_runtime.h>
#include <hip/hip_bf16.h>

typedef __attribute__((ext_vector_type(2))) float v2f;
typedef __attribute__((ext_vector_type(8))) float v8f;

#define DIM 64

// ---------------------------------------------------------------------------
// Zero the two layer buffers in workspace (must run every call: atomics
// accumulate into them and the harness does not re-poison between replays).
// ---------------------------------------------------------------------------
__global__ void zero_f4(float4* __restrict__ p, long n4) {
    long i = (long)blockIdx.x * blockDim.x + threadIdx.x;
    if (i < n4) p[i] = make_float4(0.f, 0.f, 0.f, 0.f);
}

// ---------------------------------------------------------------------------
// SpMM scatter: dst[rows[e]] += vals[e] * src[cols[e]]   (D = 64)
// 16 lanes per edge, float4 (b128) gather, 4x global_atomic_add_f32 scatter.
// Node table fits in L2 (76.8MB < 192MB) so both gather and atomics stay
// on-chip. TWO_SRC=true implements the virtual concat(user_emb, item_emb).
// ---------------------------------------------------------------------------
template <bool TWO_SRC>
__global__ void spmm_scatter(const int* __restrict__ rows,
                             const int* __restrict__ cols,
                             const float* __restrict__ vals,
                             const float* __restrict__ srcA,
                             const float* __restrict__ srcB,
                             int split,
                             float* __restrict__ dst,
                             int n_edges) {
    long tid = (long)blockIdx.x * blockDim.x + threadIdx.x;
    int e = (int)(tid >> 4);
    if (e >= n_edges) return;
    int q = ((int)tid & 15) << 2;   // dim offset, float4 granularity

    int   r = rows[e];
    int   c = cols[e];
    float v = vals[e];

    const float* s;
    if (TWO_SRC) {
        s = (c < split) ? (srcA + (long)c * DIM)
                        : (srcB + (long)(c - split) * DIM);
    } else {
        s = srcA + (long)c * DIM;
    }

    float4 x = *(const float4*)(s + q);
    float* d = dst + (long)r * DIM + q;
    atomicAdd(d + 0, v * x.x);
    atomicAdd(d + 1, v * x.y);
    atomicAdd(d + 2, v * x.z);
    atomicAdd(d + 3, v * x.w);
}

// ---------------------------------------------------------------------------
// Final stage: for each tile of 16 batch rows, build averaged embeddings
//   U[r][k] = (emb0 + l1 + l2)/3 for user rows, I[r][k] likewise for items,
// write them to d_out, and compute scores as diag(U @ I^T) with 16 chained
// V_WMMA_F32_16X16X4_F32 (K = 64 in chunks of 4, exact f32 accumulation).
//
// Fragment layout (ISA 7.12.2, 32-bit 16x4 A): lane L holds row M = L%16,
// K pair = {4*kc + 2*(L/16), +1}. B (4x16) is the mirrored layout with
// N = L%16, so the SAME per-lane float2 slices feed A (from U) and B (from I).
// C/D diag: S[m][m] lives at (lane=m, elem=m) for m<8 and (lane=m+16,
// elem=m-8) for m>=8.
// ---------------------------------------------------------------------------
__global__ void score_wmma(const int* __restrict__ users,
                           const int* __restrict__ items,
                           const float* __restrict__ user_emb,
                           const float* __restrict__ item_emb,
                           const float* __restrict__ l1,
                           const float* __restrict__ l2,
                           int n_users,
                           float* __restrict__ users_out,
                           float* __restrict__ items_out,
                           float* __restrict__ scores_out,
                           int n_tiles) {
    int lane = threadIdx.x & 31;
    int wave = threadIdx.x >> 5;
    int tile = blockIdx.x * (blockDim.x >> 5) + wave;
    if (tile >= n_tiles) return;   // whole wave exits together: EXEC all-1s

    int r = lane & 15;             // row within tile (M for A, N for B)
    int h = lane >> 4;             // K half-pair selector
    int b = tile * 16 + r;         // batch index

    int un = users[b];
    int it = items[b];
    const float* u0 = user_emb + (long)un * DIM;
    const float* u1 = l1 + (long)un * DIM;
    const float* u2 = l2 + (long)un * DIM;
    const float* i0 = item_emb + (long)it * DIM;
    const float* i1 = l1 + (long)(n_users + it) * DIM;
    const float* i2 = l2 + (long)(n_users + it) * DIM;

    const float inv3 = 1.0f / 3.0f;
    v8f acc = {};

#pragma unroll
    for (int kc = 0; kc < 16; ++kc) {
        int d = 4 * kc + 2 * h;
        v2f au, ai;
        au.x = (u0[d]     + u1[d]     + u2[d])     * inv3;
        au.y = (u0[d + 1] + u1[d + 1] + u2[d + 1]) * inv3;
        ai.x = (i0[d]     + i1[d]     + i2[d])     * inv3;
        ai.y = (i0[d + 1] + i1[d + 1] + i2[d + 1]) * inv3;

        // Averaged embeddings are exactly the fragment data: store to d_out.
        *(v2f*)(users_out + (long)b * DIM + d) = au;
        *(v2f*)(items_out + (long)b * DIM + d) = ai;

        // D = A * B + C  (A = U tile 16x4, B = I^T tile 4x16)
        acc = __builtin_amdgcn_wmma_f32_16x16x4_f32(
            /*neg_a=*/false, au, /*neg_b=*/false, ai,
            /*c_mod=*/(short)0, acc, /*reuse_a=*/false, /*reuse_b=*/false);
    }

    // Diagonal extraction -> scores (divergence only after all WMMAs).
    if (lane < 8) {
        scores_out[tile * 16 + lane] = acc[lane];
    } else if (lane >= 24) {
        scores_out[tile * 16 + (lane - 16)] = acc[lane - 24];
    }
}

// ---------------------------------------------------------------------------
// Host launcher
// ---------------------------------------------------------------------------
extern "C" void kernel_launch(void* const* d_in, const int* in_sizes, int n_in,
                              void* d_out, int out_size, void* d_ws, size_t ws_size,
                              hipStream_t stream) {
    const int*   users    = (const int*)d_in[0];
    const int*   items    = (const int*)d_in[1];
    const float* user_emb = (const float*)d_in[2];
    const float* item_emb = (const float*)d_in[3];
    const int*   rows     = (const int*)d_in[4];
    const int*   cols     = (const int*)d_in[5];
    const float* vals     = (const float*)d_in[6];

    const int batch   = in_sizes[0];
    const int n_users = in_sizes[2] / DIM;
    const int n_items = in_sizes[3] / DIM;
    const int n_nodes = n_users + n_items;
    const int n_edges = in_sizes[4];

    float* l1 = (float*)d_ws;                       // layer-1 output [n_nodes*64]
    float* l2 = l1 + (long)n_nodes * DIM;           // layer-2 output [n_nodes*64]

    // 1) zero both layer buffers (deterministic across graph replays)
    {
        long n4 = (long)n_nodes * DIM * 2 / 4;
        int bs = 256;
        unsigned nb = (unsigned)((n4 + bs - 1) / bs);
        zero_f4<<<nb, bs, 0, stream>>>((float4*)d_ws, n4);
    }

    // 2) layer 1: l1 = A_hat @ concat(user_emb, item_emb)
    {
        long nthreads = (long)n_edges * 16;
        int bs = 256;
        unsigned nb = (unsigned)((nthreads + bs - 1) / bs);
        spmm_scatter<true><<<nb, bs, 0, stream>>>(
            rows, cols, vals, user_emb, item_emb, n_users, l1, n_edges);
    }

    // 3) layer 2: l2 = A_hat @ l1
    {
        long nthreads = (long)n_edges * 16;
        int bs = 256;
        unsigned nb = (unsigned)((nthreads + bs - 1) / bs);
        spmm_scatter<false><<<nb, bs, 0, stream>>>(
            rows, cols, vals, l1, l1, 0, l2, n_edges);
    }

    // 4) gather + average + WMMA scores -> d_out = [users_emb | items_emb | scores]
    {
        float* users_out  = (float*)d_out;
        float* items_out  = users_out + (long)batch * DIM;
        float* scores_out = items_out + (long)batch * DIM;
        int n_tiles = batch / 16;                   // batch = 8192 -> 512 tiles
        int bs = 256;                               // 8 waves = 8 tiles / block
        unsigned nb = (unsigned)((n_tiles + 7) / 8);
        score_wmma<<<nb, bs, 0, stream>>>(
            users, items, user_emb, item_emb, l1, l2, n_users,
            users_out, items_out, scores_out, n_tiles);
    }
}